// DifferentiableTortuosity_68247030333964
// MI455X (gfx1250) — compile-verified
//
#include <hip/hip_runtime.h>
#include <hip/hip_bf16.h>

typedef __attribute__((ext_vector_type(2))) float v2f;
typedef __attribute__((ext_vector_type(8))) float v8f;
typedef __attribute__((ext_vector_type(4))) unsigned int v4u;
typedef __attribute__((ext_vector_type(4))) int v4i;
typedef __attribute__((ext_vector_type(8))) int v8i;

#define IMG_H   512
#define IMG_W   512
#define NBATCH  64
#define NITER   50
#define EPS     1e-6f
#define DINIT   1024.0f   // H + W

// Fused-iteration tiling: 64x32 output tile, halo 8 -> stage 80x48 (+1 zero ring)
#define OT_H 64
#define OT_W 32
#define HALO 8
#define SH   (OT_H + 2 * HALO)   // 80
#define SW   (OT_W + 2 * HALO)   // 48
#define PH   (SH + 2)            // 82 (zero ring)
#define PW   (SW + 2)            // 50
#define KPL  8                   // Jacobi iterations fused per launch

#if defined(__gfx1250__) && __has_builtin(__builtin_amdgcn_wmma_f32_16x16x4_f32)
#define USE_WMMA 1
#else
#define USE_WMMA 0
#endif

// ---- compile-probes (warnings land in stderr, rc stays 0) ------------------
#if defined(__AMDGCN__)
#if __has_include(<hip/amd_detail/amd_gfx1250_TDM.h>)
#warning "PROBE: TDM header PRESENT (expect 6-arg tensor_load_to_lds)"
#else
#warning "PROBE: TDM header ABSENT (expect 5-arg tensor_load_to_lds)"
#endif
#if __has_builtin(__builtin_amdgcn_tensor_load_to_lds)
#warning "PROBE: __builtin_amdgcn_tensor_load_to_lds PRESENT"
#else
#warning "PROBE: __builtin_amdgcn_tensor_load_to_lds ABSENT"
#endif
#if __has_builtin(__builtin_amdgcn_s_wait_tensorcnt)
#warning "PROBE: __builtin_amdgcn_s_wait_tensorcnt PRESENT"
#else
#warning "PROBE: __builtin_amdgcn_s_wait_tensorcnt ABSENT"
#endif
#if __has_builtin(__builtin_amdgcn_s_wait_asynccnt)
#warning "PROBE: __builtin_amdgcn_s_wait_asynccnt PRESENT"
#else
#warning "PROBE: __builtin_amdgcn_s_wait_asynccnt ABSENT"
#endif
#if __has_builtin(__builtin_amdgcn_global_load_async_to_lds_b32)
#warning "PROBE: __builtin_amdgcn_global_load_async_to_lds_b32 PRESENT"
#else
#warning "PROBE: __builtin_amdgcn_global_load_async_to_lds_b32 ABSENT"
#endif
#if __has_builtin(__builtin_amdgcn_global_load_async_to_lds_b128)
#warning "PROBE: __builtin_amdgcn_global_load_async_to_lds_b128 PRESENT"
#else
#warning "PROBE: __builtin_amdgcn_global_load_async_to_lds_b128 ABSENT"
#endif
#endif
// ----------------------------------------------------------------------------

#if defined(__gfx1250__) && __has_builtin(__builtin_amdgcn_tensor_load_to_lds)
#define USE_TDM 1
#if __has_include(<hip/amd_detail/amd_gfx1250_TDM.h>)
#define TDM_6ARG 1
#else
#define TDM_6ARG 0
#endif
#else
#define USE_TDM 0
#endif

__device__ __forceinline__ int iabs_(int x) { return x < 0 ? -x : x; }
__device__ __forceinline__ int iclamp_(int x, int lo, int hi) {
  return x < lo ? lo : (x > hi ? hi : x);
}

// ---------------------------------------------------------------------------
__global__ __launch_bounds__(256) void tort_init(float* __restrict__ D, int n) {
  int i = blockIdx.x * blockDim.x + threadIdx.x;
  if (i < n) D[i] = DINIT;
}

__global__ void tort_set_goal(const int* __restrict__ goal, float* __restrict__ D) {
  int b = threadIdx.x;
  if (b < NBATCH) {
    int gr = iclamp_(goal[2 * b + 0], 0, IMG_H - 1);
    int gc = iclamp_(goal[2 * b + 1], 0, IMG_W - 1);
    D[(size_t)b * IMG_H * IMG_W + (size_t)gr * IMG_W + gc] = 0.0f;
  }
}

// ---------------------------------------------------------------------------
#if USE_TDM
// One TDM descriptor: nrows iterations of a 1-row tile of `width` fp32
// elements; LDS advances by PW elements per row, global by IMG_W elements.
// Right-edge zero fill via tensor_dim0 = elements left in the image row.
__device__ __forceinline__ void tdm_load_rows(unsigned lds_byte_addr,
                                              const float* gsrc,
                                              int nrows, int width,
                                              int tensor_d0) {
  unsigned long long ga = (unsigned long long)(uintptr_t)gsrc;
  v4u g0;
  g0.x = 1u;                                              // count=1, user mode
  g0.y = lds_byte_addr;                                   // lds_addr (bytes)
  g0.z = (unsigned)(ga & 0xFFFFFFFFu);                    // global_addr lo
  g0.w = (unsigned)((ga >> 32) & 0x01FFFFFFu) | (2u << 30); // addr hi | type=2

  unsigned td0 = (unsigned)tensor_d0;
  v8i g1;
  g1[0] = (int)((2u << 16) | (1u << 19));                 // data_size=4B, iterate
  g1[1] = (int)((td0 & 0xFFFFu) << 16);                   // tensor_dim0[15:0]
  g1[2] = (int)(((td0 >> 16) & 0xFFFFu) | ((unsigned)IMG_H << 16)); // td0 hi | td1 lo
  g1[3] = (int)(((unsigned)width & 0xFFFFu) << 16);       // td1 hi(0) | tile_dim0
  g1[4] = 1;                                              // tile_dim1=1, tile_dim2=0
  g1[5] = IMG_W;                                          // tensor_dim0_stride lo
  g1[6] = 0;
  g1[7] = 0;

  v4i g2;
  g2[0] = 1;                                              // tensor_dim2 (unused)
  g2[1] = PW;                                             // lds_addr_increment (elems)
  g2[2] = IMG_W;                                          // global_addr_increment lo
  g2[3] = (int)(((unsigned)(nrows - 1) & 0xFFFFu) << 16); // iterate_count
  v4i g3 = {0, 0, 0, 0};

#if TDM_6ARG
  v8i g4 = {0, 0, 0, 0, 0, 0, 0, 0};   // extra group on clang-23 form (zeros)
  __builtin_amdgcn_tensor_load_to_lds(g0, g1, g2, g3, g4, 0);
#else
  __builtin_amdgcn_tensor_load_to_lds(g0, g1, g2, g3, 0);
#endif
}
#endif

// ---------------------------------------------------------------------------
// Fused sweep kernel: stage D,p tile (+halo) -> niter Jacobi iterations in LDS
// -> write exact 64x32 interior. Neighbor sums per 16x16 subtile via 8 chained
// V_WMMA_F32_16X16X4_F32 (S*D + D*S, S = |i-j|==1 indicator). All LDS operands
// are gathered BEFORE the WMMA chain so there is a single dscnt wait-front and
// the matrix pipe overlaps the load tail.
__global__ __launch_bounds__(256) void tort_fused(const float* __restrict__ p,
                                                  const float* __restrict__ Din,
                                                  float* __restrict__ Dout,
                                                  int niter) {
  __shared__ float Dsh[2][PH * PW];
  __shared__ float Ish[PH * PW];

  const int tid  = threadIdx.x;
  const int wave = tid >> 5;
  const int lane = tid & 31;
  const int m    = lane & 15;
  const int hi   = lane >> 4;

  const int wg   = blockIdx.x;                 // 64 batches * 8 * 16 tiles
  const int b    = wg >> 7;
  const int t    = wg & 127;
  const int row0 = (t >> 4) * OT_H;
  const int col0 = (t & 15) * OT_W;
  const int gr0  = row0 - HALO;
  const int gc0  = col0 - HALO;

  const size_t base = (size_t)b * (IMG_H * IMG_W);
  const float* __restrict__ Dg = Din + base;
  const float* __restrict__ Pg = p   + base;

  // Zero padded buffers: zero ring + out-of-image cells act as zero padding.
  for (int i = tid; i < PH * PW; i += 256) {
    Dsh[0][i] = 0.0f;
    Dsh[1][i] = 0.0f;
    Ish[i]    = 0.0f;
  }
  __syncthreads();

  // Clamped intersection of the staged region with the image.
  const int r_lo    = gr0 < 0 ? 0 : gr0;
  const int c_lo    = gc0 < 0 ? 0 : gc0;
  const int row_off = r_lo - gr0;
  const int col_off = c_lo - gc0;
  const int r_hi    = (gr0 + SH) > IMG_H ? IMG_H : (gr0 + SH);
  const int nrows   = r_hi - r_lo;
  const int width   = SW - col_off;

#if USE_TDM
  if (wave == 0) {
    unsigned ldsD = (unsigned)(uintptr_t)&Dsh[0][(1 + row_off) * PW + (1 + col_off)];
    unsigned ldsI = (unsigned)(uintptr_t)&Ish[(1 + row_off) * PW + (1 + col_off)];
    const float* gD = &Dg[(size_t)r_lo * IMG_W + c_lo];
    const float* gP = &Pg[(size_t)r_lo * IMG_W + c_lo];
    tdm_load_rows(ldsD, gD, nrows, width, IMG_W - c_lo);
    tdm_load_rows(ldsI, gP, nrows, width, IMG_W - c_lo);
#if __has_builtin(__builtin_amdgcn_s_wait_tensorcnt)
    __builtin_amdgcn_s_wait_tensorcnt((short)0);
#else
    asm volatile("s_wait_tensorcnt 0x0" ::: "memory");
#endif
  }
#else
  for (int idx = tid; idx < SH * SW; idx += 256) {
    const int i  = idx / SW;
    const int j  = idx - i * SW;
    const int gr = gr0 + i;
    const int gc = gc0 + j;
    if (gr >= 0 && gr < IMG_H && gc >= 0 && gc < IMG_W) {
      const size_t g = (size_t)gr * IMG_W + gc;
      Dsh[0][(i + 1) * PW + (j + 1)] = Dg[g];
      Ish[(i + 1) * PW + (j + 1)]    = Pg[g];
    }
  }
#endif
  __syncthreads();

  // p -> 1/(p+eps). Value at out-of-image cells is irrelevant: D there is 0
  // and min(0, eff>=0) keeps it 0.
  for (int i = tid; i < PH * PW; i += 256)
    Ish[i] = 1.0f / (Ish[i] + EPS);
  __syncthreads();

#if USE_WMMA
  // Constant S operand chunks (pure function of lane).
  v2f sOp[4];
  #pragma unroll
  for (int q = 0; q < 4; ++q) {
    const int k0 = 4 * q + 2 * hi;
    sOp[q].x = (iabs_(m - k0) == 1) ? 1.0f : 0.0f;
    sOp[q].y = (iabs_(m - (k0 + 1)) == 1) ? 1.0f : 0.0f;
  }
#endif

  // Fused Jacobi iterations, ping-pong in LDS. 15 subtiles (5x3 of 16x16)
  // over the 80x48 core, 8 waves -> two passes (wave-uniform predicates).
  for (int it = 0; it < niter; ++it) {
    const int src = it & 1;
    const int dst = src ^ 1;
    for (int pass = 0; pass < 2; ++pass) {
      const int s = wave + (pass << 3);
      if (s < 15) {
        const int srow = (s / 3) << 4;
        const int scol = (s - (s / 3) * 3) << 4;
        const int O    = (srow + 1) * PW + (scol + 1);   // padded origin

        // ---- gather ALL operands first (single dscnt wait-front) ----
        float ctr[8], inv[8], lf[8], rt[8];
#if USE_WMMA
        v2f aD[4], bD[4];
        #pragma unroll
        for (int q = 0; q < 4; ++q) {
          const int k0 = 4 * q + 2 * hi;
          aD[q].x = Dsh[src][O + m * PW + k0];
          aD[q].y = Dsh[src][O + m * PW + k0 + 1];
          bD[q].x = Dsh[src][O + k0 * PW + m];
          bD[q].y = Dsh[src][O + (k0 + 1) * PW + m];
        }
        const float topv = Dsh[src][O - PW + m];
        const float botv = Dsh[src][O + 16 * PW + m];
#endif
        #pragma unroll
        for (int r = 0; r < 8; ++r) {
          const int row = O + (r + 8 * hi) * PW;
          lf[r]  = Dsh[src][row - 1];
          rt[r]  = Dsh[src][row + 16];
          ctr[r] = Dsh[src][row + m];
          inv[r] = Ish[row + m];
        }

        v8f c = {0.f, 0.f, 0.f, 0.f, 0.f, 0.f, 0.f, 0.f};
#if USE_WMMA
        #pragma unroll
        for (int q = 0; q < 4; ++q)   // vertical sums: S * D
          c = __builtin_amdgcn_wmma_f32_16x16x4_f32(false, sOp[q], false, bD[q],
                                                    (short)0, c, false, false);
        #pragma unroll
        for (int q = 0; q < 4; ++q)   // horizontal sums: D * S
          c = __builtin_amdgcn_wmma_f32_16x16x4_f32(false, aD[q], false, sOp[q],
                                                    (short)0, c, false, false);
        c[0] += (hi == 0) ? topv : 0.0f;   // output row M=0
        c[7] += (hi == 1) ? botv : 0.0f;   // output row M=15
#else
        #pragma unroll
        for (int r = 0; r < 8; ++r) {
          const int row = O + (r + 8 * hi) * PW;
          c[r] = Dsh[src][row - PW + m] + Dsh[src][row + PW + m];
        }
        // lf/rt below add the horizontal pair only at tile edges in the WMMA
        // path; in the fallback add interior horizontal neighbors directly.
        #pragma unroll
        for (int r = 0; r < 8; ++r) {
          const int row = O + (r + 8 * hi) * PW;
          c[r] += Dsh[src][row + m - 1] + Dsh[src][row + m + 1];
        }
#endif
        // ---- combine + relax + store ----
        #pragma unroll
        for (int r = 0; r < 8; ++r) {
#if USE_WMMA
          float h = (m == 0)  ? lf[r] : 0.0f;
          h       = (m == 15) ? rt[r] : h;
          const float nb = c[r] + h;
#else
          const float nb = c[r];
#endif
          const float eff = (0.25f * nb + 1.0f) * inv[r];
          Dsh[dst][O + (r + 8 * hi) * PW + m] = fminf(ctr[r], eff);
        }
      }
    }
    __syncthreads();
  }

  // Write back the exact interior (distance >= HALO from staged border).
  const int fin = niter & 1;
  for (int idx = tid; idx < OT_H * OT_W; idx += 256) {
    const int i = idx >> 5;
    const int j = idx & 31;
    Dout[base + (size_t)(row0 + i) * IMG_W + (col0 + j)] =
        Dsh[fin][(HALO + 1 + i) * PW + (HALO + 1 + j)];
  }
}

// ---------------------------------------------------------------------------
__global__ void tort_finish(const float* __restrict__ Dfin,
                            const int* __restrict__ start,
                            const int* __restrict__ goal,
                            float* __restrict__ out, int out_size) {
  int b = threadIdx.x;
  if (b < NBATCH) {
    int sr = iclamp_(start[2 * b + 0], 0, IMG_H - 1);
    int sc = iclamp_(start[2 * b + 1], 0, IMG_W - 1);
    float path = Dfin[(size_t)b * IMG_H * IMG_W + (size_t)sr * IMG_W + sc];
    float dr = (float)(goal[2 * b + 0] - start[2 * b + 0]);
    float dc = (float)(goal[2 * b + 1] - start[2 * b + 1]);
    float e  = sqrtf(dr * dr + dc * dc);
    e = fmaxf(e, 1.0f);
    out[b] = path / e;
    if (out_size >= 2 * NBATCH)
      out[NBATCH + b] = (path < DINIT) ? 1.0f : 0.0f;
  }
}

// ---------------------------------------------------------------------------
extern "C" void kernel_launch(void* const* d_in, const int* in_sizes, int n_in,
                              void* d_out, int out_size, void* d_ws, size_t ws_size,
                              hipStream_t stream) {
  (void)in_sizes; (void)n_in; (void)ws_size;
  const float* p     = (const float*)d_in[0];
  const int*   start = (const int*)d_in[1];
  const int*   goal  = (const int*)d_in[2];
  float*       out   = (float*)d_out;

  const size_t plane = (size_t)NBATCH * IMG_H * IMG_W;
  float* Da = (float*)d_ws;            // ping
  float* Db = Da + plane;              // pong  (needs 128 MB of d_ws)

  tort_init<<<(int)((plane + 255) / 256), 256, 0, stream>>>(Da, (int)plane);
  tort_set_goal<<<1, 64, 0, stream>>>(goal, Da);

  const int blocks = NBATCH * (IMG_H / OT_H) * (IMG_W / OT_W);  // 8192
  static const int counts[7] = {KPL, KPL, KPL, KPL, KPL, KPL, NITER - 6 * KPL};
  float* cur = Da;
  float* nxt = Db;
  for (int l = 0; l < 7; ++l) {
    tort_fused<<<blocks, 256, 0, stream>>>(p, cur, nxt, counts[l]);
    float* tmp = cur; cur = nxt; nxt = tmp;
  }

  tort_finish<<<1, 64, 0, stream>>>(cur, start, goal, out, out_size);
}